// GraphConv_2353642078695
// MI455X (gfx1250) — compile-verified
//
#include <hip/hip_runtime.h>
#include <math.h>

typedef __attribute__((ext_vector_type(2))) float v2f;
typedef __attribute__((ext_vector_type(4))) float v4f;
typedef __attribute__((ext_vector_type(8))) float v8f;

// ---------------------------------------------------------------------------
// K1: zero the output (used as the rst accumulator) and the degree buffer,
//     and pre-swizzle W[128][64] into WMMA B-fragment order:
//     frag (tile,kk,lane) -> v2f{ W[4*kk+2*hi][16*tile + (lane&15)],
//                                 W[4*kk+2*hi+1][...] },  hi = lane>>4
// ---------------------------------------------------------------------------
__global__ void k_init(float* __restrict__ out, float* __restrict__ deg,
                       float* __restrict__ wf, const float* __restrict__ W,
                       int n) {
    int idx = blockIdx.x * blockDim.x + threadIdx.x;
    int total = n * 64;
    if (idx < total) out[idx] = 0.0f;
    if (idx < n)     deg[idx] = 0.0f;
    if (idx < 4096) {                 // 4 tiles * 32 ksteps * 32 lanes
        int L    = idx & 31;
        int f    = idx >> 5;
        int kk   = f & 31;
        int tile = f >> 5;
        int hi   = L >> 4;
        int col  = tile * 16 + (L & 15);
        int k    = kk * 4 + 2 * hi;
        wf[2 * idx + 0] = W[k * 64 + col];
        wf[2 * idx + 1] = W[(k + 1) * 64 + col];
    }
}

// ---------------------------------------------------------------------------
// K2: masked in-degree:  deg[dst[e]] += mask[src[e]]
// ---------------------------------------------------------------------------
__global__ void k_degree(const int* __restrict__ src, const int* __restrict__ dst,
                         const int* __restrict__ mask, float* __restrict__ deg,
                         int e) {
    int i = blockIdx.x * blockDim.x + threadIdx.x;
    if (i >= e) return;
    int s = src[i];
    if (mask[s]) atomicAdd(&deg[dst[i]], 1.0f);
}

// ---------------------------------------------------------------------------
// K3: norm[i] = deg==0 ? 1 : deg^-0.5
// ---------------------------------------------------------------------------
__global__ void k_norm(const float* __restrict__ deg, float* __restrict__ nrm,
                       int n) {
    int i = blockIdx.x * blockDim.x + threadIdx.x;
    if (i >= n) return;
    float d = deg[i];
    nrm[i] = (d == 0.0f) ? 1.0f : (1.0f / sqrtf(d));
}

// ---------------------------------------------------------------------------
// K4: edge aggregation.  16 threads per edge, 4 floats each (float4 gather).
//     rst[dst] += feat[src] * norm[src]   (skipped entirely when mask[src]==0)
//     All traffic is L2-resident (feat+rst ~51MB << 192MB L2).
// ---------------------------------------------------------------------------
__global__ void k_scatter(const float* __restrict__ feat, const int* __restrict__ src,
                          const int* __restrict__ dst, const int* __restrict__ mask,
                          const float* __restrict__ nrm, float* __restrict__ rst,
                          int e) {
    long long t = (long long)blockIdx.x * blockDim.x + threadIdx.x;
    int edge = (int)(t >> 4);
    if (edge >= e) return;
    int s = src[edge];
    if (!mask[s]) return;                       // h is exactly zero -> no atomics
    int d  = dst[edge];
    float nm = nrm[s];
    int c4 = (int)(t & 15);
    v4f v = ((const v4f*)(feat + (size_t)s * 64))[c4];
    float* r = rst + (size_t)d * 64 + c4 * 4;
    atomicAdd(r + 0, v.x * nm);
    atomicAdd(r + 1, v.y * nm);
    atomicAdd(r + 2, v.z * nm);
    atomicAdd(r + 3, v.w * nm);
}

// ---------------------------------------------------------------------------
// K5: fused  out = concat(rst*norm, feat) @ W + bias  via V_WMMA_F32_16X16X4_F32
//     One wave per 16-row tile; A fragments (32 x v2f) held in registers and
//     reused across the 4 column tiles.  rst lives in `out` (read-before-write
//     within the owning wave only).
// ---------------------------------------------------------------------------
__global__ __launch_bounds__(256) void k_gemm(const float* __restrict__ rst,
                                              const float* __restrict__ feat,
                                              const float* __restrict__ nrm,
                                              const float* __restrict__ wf,
                                              const float* __restrict__ bias,
                                              float* __restrict__ out, int n) {
    int lane  = threadIdx.x & 31;
    int wave  = threadIdx.x >> 5;
    int tile16 = blockIdx.x * 8 + wave;
    int ntiles = (n + 15) >> 4;
    if (tile16 >= ntiles) return;               // whole-wave exit: EXEC stays all-1s

    int base = tile16 * 16;
    int hi   = lane >> 4;
    int rl   = lane & 15;
    int row  = base + rl;
    if (row >= n) row = n - 1;                  // clamp (N divisible by 16 anyway)

    float nm = nrm[row];
    const float* rrow = rst  + (size_t)row * 64;
    const float* frow = feat + (size_t)row * 64;

    // A fragments: lane holds X[row][4*kk + 2*hi .. +1], X = [rst*norm | feat]
    v2f a[32];
#pragma unroll
    for (int kk = 0; kk < 32; ++kk) {
        int k = kk * 4 + 2 * hi;
        if (k < 64) {
            a[kk].x = rrow[k]     * nm;
            a[kk].y = rrow[k + 1] * nm;
        } else {
            a[kk].x = frow[k - 64];
            a[kk].y = frow[k - 63];
        }
    }

    const v2f* wf2 = (const v2f*)wf;
#pragma unroll
    for (int tile = 0; tile < 4; ++tile) {
        v8f acc = {0.f, 0.f, 0.f, 0.f, 0.f, 0.f, 0.f, 0.f};
#pragma unroll
        for (int kk = 0; kk < 32; ++kk) {
            v2f b = wf2[(tile * 32 + kk) * 32 + lane];
            acc = __builtin_amdgcn_wmma_f32_16x16x4_f32(
                false, a[kk], false, b, (short)0, acc, false, false);
        }
        int col = tile * 16 + rl;
        float bs = bias[col];
#pragma unroll
        for (int v = 0; v < 8; ++v) {
            int orow = base + v + 8 * hi;       // D VGPR v -> row v (+8 for hi lanes)
            if (orow < n) out[(size_t)orow * 64 + col] = acc[v] + bs;
        }
    }
}

// ---------------------------------------------------------------------------
extern "C" void kernel_launch(void* const* d_in, const int* in_sizes, int n_in,
                              void* d_out, int out_size, void* d_ws, size_t ws_size,
                              hipStream_t stream) {
    const float* feat = (const float*)d_in[0];
    const int*   src  = (const int*)  d_in[1];
    const int*   dst  = (const int*)  d_in[2];
    const int*   mask = (const int*)  d_in[3];
    const float* W    = (const float*)d_in[4];
    const float* bias = (const float*)d_in[5];
    float* out = (float*)d_out;

    int n = in_sizes[3];   // mask has N elements
    int e = in_sizes[1];   // src has E elements

    float* deg = (float*)d_ws;     // N floats
    float* nrm = deg + n;          // N floats
    float* wf  = nrm + n;          // 8192 floats (pre-swizzled W fragments)

    int total = n * 64;
    k_init  <<<(total + 255) / 256, 256, 0, stream>>>(out, deg, wf, W, n);
    k_degree<<<(e + 255) / 256,     256, 0, stream>>>(src, dst, mask, deg, e);
    k_norm  <<<(n + 255) / 256,     256, 0, stream>>>(deg, nrm, n);

    long long st = (long long)e * 16;
    k_scatter<<<(unsigned)((st + 255) / 256), 256, 0, stream>>>(feat, src, dst, mask,
                                                                nrm, out, e);

    int ntiles = (n + 15) >> 4;
    k_gemm<<<(ntiles + 7) / 8, 256, 0, stream>>>(out, feat, nrm, wf, bias, out, n);
}